// bce_45045617001061
// MI455X (gfx1250) — compile-verified
//
#include <hip/hip_runtime.h>

typedef __attribute__((ext_vector_type(2))) float v2f;
typedef __attribute__((ext_vector_type(4))) float f32x4;
typedef __attribute__((ext_vector_type(4))) int   i32x4;
typedef __attribute__((ext_vector_type(8))) float v8f;

// Full wave32 sum via V_WMMA_F32_16X16X4_F32.
// A (16x4, 2 VGPRs): lane L<16 -> A[L][0]=s (VGPR0), A[L][1]=0 (VGPR1);
//                    lane L>=16 -> A[L-16][2]=s, A[L-16][3]=0.
// B = all ones (layout independent). D[m][n] = sum_k A[m][k] = s_m + s_{m+16}.
// D VGPR j holds rows M=j (lanes 0-15) and M=j+8 (lanes 16-31), so summing the
// 8 D VGPRs gives sum over M=0..7 (low lanes) / M=8..15 (high lanes); one
// xor-16 shuffle-add completes the 32-lane reduction. EXEC must be all ones
// at the call site (callers reconverge before calling).
__device__ __forceinline__ float wave_sum_wmma(float s) {
    v2f a; a[0] = s;    a[1] = 0.0f;
    v2f b; b[0] = 1.0f; b[1] = 1.0f;
    v8f c = {};
    v8f d = __builtin_amdgcn_wmma_f32_16x16x4_f32(
        /*neg_a=*/false, a, /*neg_b=*/false, b,
        /*c_mod=*/(short)0, c, /*reuse_a=*/false, /*reuse_b=*/false);
    float t = ((d[0] + d[1]) + (d[2] + d[3])) + ((d[4] + d[5]) + (d[6] + d[7]));
    t += __shfl_xor(t, 16, 32);
    return t;
}

// Block (256 threads = 8 wave32) sum; result valid on thread 0.
__device__ __forceinline__ float block_sum(float s) {
    __shared__ float smem[8];
    float w = wave_sum_wmma(s);
    const int lane = threadIdx.x & 31;
    const int wid  = threadIdx.x >> 5;
    if (lane == 0) smem[wid] = w;
    __syncthreads();
    float r = 0.0f;
    if (threadIdx.x == 0) {
#pragma unroll
        for (int i = 0; i < 8; ++i) r += smem[i];
    }
    return r;
}

__global__ __launch_bounds__(256) void bce_partial_kernel(
    const f32x4* __restrict__ p4, const i32x4* __restrict__ t4,
    const float* __restrict__ ps, const int* __restrict__ ts,
    float* __restrict__ partial, int nvec, int ntail) {
    float acc = 0.0f;  // accumulates log2 terms; scaled by ln2 in pass 2
    int idx    = blockIdx.x * 256 + threadIdx.x;
    int stride = gridDim.x * 256;
    for (int i = idx; i < nvec; i += stride) {
        // streamed once, bigger than L2 -> non-temporal b128 loads
        f32x4 p = __builtin_nontemporal_load(p4 + i);
        i32x4 t = __builtin_nontemporal_load(t4 + i);
#pragma unroll
        for (int j = 0; j < 4; ++j) {
            float q = (t[j] == 0) ? (1.0f - p[j]) : p[j];
            acc += __builtin_amdgcn_logf(q);   // v_log_f32 (log2)
        }
    }
    // scalar tail (n % 4), handled by a single thread, then reconverge
    if (blockIdx.x == 0 && threadIdx.x == 0) {
        const int base = nvec * 4;
        for (int k = 0; k < ntail; ++k) {
            float pv = ps[base + k];
            int   tv = ts[base + k];
            float q  = (tv == 0) ? (1.0f - pv) : pv;
            acc += __builtin_amdgcn_logf(q);
        }
    }
    float bs = block_sum(acc);   // EXEC all ones here: full 256-thread block
    if (threadIdx.x == 0) partial[blockIdx.x] = bs;
}

__global__ __launch_bounds__(256) void bce_final_kernel(
    const float* __restrict__ partial, int nparts,
    float* __restrict__ out, float scale) {
    float acc = 0.0f;
    for (int i = threadIdx.x; i < nparts; i += 256) acc += partial[i];
    float bs = block_sum(acc);
    if (threadIdx.x == 0) out[0] = bs * scale;   // scale = -ln2 / N
}

extern "C" void kernel_launch(void* const* d_in, const int* in_sizes, int n_in,
                              void* d_out, int out_size, void* d_ws, size_t ws_size,
                              hipStream_t stream) {
    const float* p = (const float*)d_in[0];
    const int*   t = (const int*)d_in[1];
    const long long n = (long long)in_sizes[0];   // 16384*4096 = 67,108,864
    const int nvec  = (int)(n / 4);
    const int ntail = (int)(n % 4);

    float* partial = (float*)d_ws;                // 2048 floats = 8 KB scratch
    const int GRID = 2048;                        // 16384 wave32 in flight

    bce_partial_kernel<<<GRID, 256, 0, stream>>>(
        (const f32x4*)p, (const i32x4*)t, p, t, partial, nvec, ntail);

    const float scale = (float)(-0.69314718055994530942 / (double)n);
    bce_final_kernel<<<1, 256, 0, stream>>>(partial, GRID, (float*)d_out, scale);
}